// GNN_3324304687820
// MI455X (gfx1250) — compile-verified
//
#include <hip/hip_runtime.h>

typedef float v2f __attribute__((ext_vector_type(2)));
typedef float v8f __attribute__((ext_vector_type(8)));

#define HDIM 32
#define NEG_SLOPE 0.01f

__device__ __forceinline__ float lrelu(float v) { return v > 0.0f ? v : NEG_SLOPE * v; }

// ---------------------------------------------------------------------------
// Zero-fill (grid-stride)
// ---------------------------------------------------------------------------
__global__ void k_zero(float* __restrict__ p, long n) {
    long i = (long)blockIdx.x * blockDim.x + threadIdx.x;
    long stride = (long)gridDim.x * blockDim.x;
    for (; i < n; i += stride) p[i] = 0.0f;
}

// ---------------------------------------------------------------------------
// xu gather: ws[i][:] = emb[ids[i]][:], also write final out cols [0,32)
// one thread per (row, col)
// ---------------------------------------------------------------------------
__global__ void k_gather(const float* __restrict__ emb, const int* __restrict__ ids,
                         long n, float* __restrict__ ws, float* __restrict__ fin) {
    long t = (long)blockIdx.x * blockDim.x + threadIdx.x;
    if (t >= n * HDIM) return;
    long i = t >> 5;
    int  c = (int)(t & 31);
    float v = emb[(long)ids[i] * HDIM + c];
    ws[i * HDIM + c]  = v;
    fin[i * 96 + c]   = v;
}

// ---------------------------------------------------------------------------
// xp = item_emb[pid] + X @ W^T + b        (X: [n,D], W: [32,D])
// one wave per 16-row tile, fp32 WMMA 16x16x4, K-loop over D
// ---------------------------------------------------------------------------
__global__ void k_feat_gemm(const float* __restrict__ X, const float* __restrict__ W,
                            const float* __restrict__ b, const float* __restrict__ emb,
                            const int* __restrict__ pid, int n, int D,
                            float* __restrict__ ws, float* __restrict__ fin) {
    int wave = blockIdx.x * (blockDim.x >> 5) + (threadIdx.x >> 5);
    int lane = threadIdx.x & 31;
    int ntiles = (n + 15) >> 4;
    if (wave >= ntiles) return;              // wave-uniform: EXEC stays full
    int i0 = wave << 4;
    int m  = lane & 15;
    int hh = lane >> 4;
    int arow = i0 + m; if (arow >= n) arow = n - 1;
    const float* Arow = X + (long)arow * D;
    const float* B0   = W + (long)m * D;           // output col n = lane&15
    const float* B1   = W + (long)(16 + m) * D;    // output col 16+n

    v8f c0 = {}; v8f c1 = {};
    for (int kk = 0; kk < D; kk += 4) {
        int k2 = kk + 2 * hh;
        v2f a  = *(const v2f*)(Arow + k2);
        v2f b0 = *(const v2f*)(B0 + k2);
        v2f b1 = *(const v2f*)(B1 + k2);
        c0 = __builtin_amdgcn_wmma_f32_16x16x4_f32(false, a, false, b0, (short)0, c0, false, false);
        c1 = __builtin_amdgcn_wmma_f32_16x16x4_f32(false, a, false, b1, (short)0, c1, false, false);
    }
    int ncol0 = m, ncol1 = 16 + m;
    float bias0 = b[ncol0], bias1 = b[ncol1];
    for (int r = 0; r < 8; ++r) {
        int row = i0 + r + 8 * hh;
        if (row < n) {
            long e = (long)pid[row] * HDIM;
            float v0 = c0[r] + bias0 + emb[e + ncol0];
            float v1 = c1[r] + bias1 + emb[e + ncol1];
            ws[(long)row * HDIM + ncol0] = v0;
            ws[(long)row * HDIM + ncol1] = v1;
            fin[(long)row * 96 + ncol0]  = v0;
            fin[(long)row * 96 + ncol1]  = v1;
        }
    }
}

// ---------------------------------------------------------------------------
// Edge scatter: msg[dst][c] += xsrc[src][c]; cnt[dst] += 1
// one thread per (edge, col)
// ---------------------------------------------------------------------------
__global__ void k_scatter(const float* __restrict__ xsrc, const int* __restrict__ esrc,
                          const int* __restrict__ edst, long E,
                          float* __restrict__ msg, float* __restrict__ cnt) {
    long t = (long)blockIdx.x * blockDim.x + threadIdx.x;
    if (t >= E * HDIM) return;
    long e = t >> 5;
    int  c = (int)(t & 31);
    int s = esrc[e];
    int d = edst[e];
    atomicAdd(msg + (long)d * HDIM + c, xsrc[(long)s * HDIM + c]);
    if (c == 0) atomicAdd(cnt + d, 1.0f);
}

// ---------------------------------------------------------------------------
// Fused SAGE epilogue: out = lrelu( (msg/max(cnt,1)) @ Wl^T + bl + xdst @ Wr^T )
// one wave per 16-row tile; single K=64 GEMM over [mean | xdst] @ [Wl | Wr]^T
// writes ws (stride 32, optional) and final out slice (stride 96)
// ---------------------------------------------------------------------------
__global__ void k_sage(const float* __restrict__ msg, const float* __restrict__ cnt,
                       const float* __restrict__ xdst,
                       const float* __restrict__ Wl, const float* __restrict__ bl,
                       const float* __restrict__ Wr,
                       int n, float* __restrict__ ws, float* __restrict__ fin) {
    int wave = blockIdx.x * (blockDim.x >> 5) + (threadIdx.x >> 5);
    int lane = threadIdx.x & 31;
    int ntiles = (n + 15) >> 4;
    if (wave >= ntiles) return;              // wave-uniform
    int i0 = wave << 4;
    int m  = lane & 15;
    int hh = lane >> 4;
    int arow = i0 + m; if (arow >= n) arow = n - 1;
    float ic = 1.0f / fmaxf(cnt[arow], 1.0f);       // hoisted mean divisor
    const float* Ml = msg  + (long)arow * HDIM;
    const float* Xd = xdst + (long)arow * HDIM;

    v8f c0 = {}; v8f c1 = {};
    for (int kk = 0; kk < 64; kk += 4) {
        int k2 = (kk & 31) + 2 * hh;
        v2f a, b0, b1;
        if (kk < 32) {
            v2f mm = *(const v2f*)(Ml + k2);
            a.x = mm.x * ic; a.y = mm.y * ic;
            b0 = *(const v2f*)(Wl + (long)m * HDIM + k2);
            b1 = *(const v2f*)(Wl + (long)(16 + m) * HDIM + k2);
        } else {
            a  = *(const v2f*)(Xd + k2);
            b0 = *(const v2f*)(Wr + (long)m * HDIM + k2);
            b1 = *(const v2f*)(Wr + (long)(16 + m) * HDIM + k2);
        }
        c0 = __builtin_amdgcn_wmma_f32_16x16x4_f32(false, a, false, b0, (short)0, c0, false, false);
        c1 = __builtin_amdgcn_wmma_f32_16x16x4_f32(false, a, false, b1, (short)0, c1, false, false);
    }
    float bias0 = bl[m], bias1 = bl[16 + m];
    for (int r = 0; r < 8; ++r) {
        int row = i0 + r + 8 * hh;
        if (row < n) {
            float v0 = lrelu(c0[r] + bias0);
            float v1 = lrelu(c1[r] + bias1);
            if (ws) {
                ws[(long)row * HDIM + m]      = v0;
                ws[(long)row * HDIM + 16 + m] = v1;
            }
            fin[(long)row * 96 + m]      = v0;
            fin[(long)row * 96 + 16 + m] = v1;
        }
    }
}

// ---------------------------------------------------------------------------
extern "C" void kernel_launch(void* const* d_in, const int* in_sizes, int n_in,
                              void* d_out, int out_size, void* d_ws, size_t ws_size,
                              hipStream_t stream) {
    const int*   user_ids     = (const int*)d_in[0];
    const int*   product_ids  = (const int*)d_in[1];
    const float* product_feat = (const float*)d_in[2];
    const int*   edge_rates   = (const int*)d_in[3];   // [2, E]: row0=user(src), row1=item(dst)
    const int*   edge_rated   = (const int*)d_in[4];   // [2, E]: row0=item(src), row1=user(dst)
    const float* user_emb     = (const float*)d_in[5];
    const float* item_emb     = (const float*)d_in[6];
    const float* feat_W       = (const float*)d_in[7];
    const float* feat_b       = (const float*)d_in[8];
    const float* Wl_up1 = (const float*)d_in[9],  *bl_up1 = (const float*)d_in[10], *Wr_up1 = (const float*)d_in[11];
    const float* Wl_pu1 = (const float*)d_in[12], *bl_pu1 = (const float*)d_in[13], *Wr_pu1 = (const float*)d_in[14];
    const float* Wl_up2 = (const float*)d_in[15], *bl_up2 = (const float*)d_in[16], *Wr_up2 = (const float*)d_in[17];
    const float* Wl_pu2 = (const float*)d_in[18], *bl_pu2 = (const float*)d_in[19], *Wr_pu2 = (const float*)d_in[20];

    const long Nu = in_sizes[0];
    const long Ni = in_sizes[1];
    const long E  = (long)in_sizes[3] / 2;
    const int  D  = in_sizes[7] / HDIM;    // 256
    const long maxN = Nu > Ni ? Nu : Ni;

    // workspace layout (floats)
    float* ws  = (float*)d_ws;
    float* xu  = ws;                 // [Nu,32]
    float* xp  = xu  + Nu * HDIM;    // [Ni,32]
    float* u1  = xp  + Ni * HDIM;    // [Nu,32]
    float* p1  = u1  + Nu * HDIM;    // [Ni,32]
    float* msg = p1  + Ni * HDIM;    // [maxN,32]
    float* cnt = msg + maxN * HDIM;  // [maxN]

    float* out      = (float*)d_out;
    float* out_user = out;            // [Nu,96]
    float* out_item = out + Nu * 96;  // [Ni,96]

    const int  TB = 256;
    const int  WPB = TB >> 5;                         // 8 waves/block for GEMM kernels
    const long gGather  = (Nu * HDIM + TB - 1) / TB;
    const long gScatter = (E * HDIM + TB - 1) / TB;
    const int  tilesU = (int)((Nu + 15) >> 4), gU = (tilesU + WPB - 1) / WPB;
    const int  tilesI = (int)((Ni + 15) >> 4), gI = (tilesI + WPB - 1) / WPB;

    // xu = user_emb[user_ids]  -> ws + out_user[:, 0:32]
    k_gather<<<(unsigned)gGather, TB, 0, stream>>>(user_emb, user_ids, Nu, xu, out_user);
    // xp = item_emb[product_ids] + feat @ W^T + b -> ws + out_item[:, 0:32]
    k_feat_gemm<<<(unsigned)gI, TB, 0, stream>>>(product_feat, feat_W, feat_b, item_emb,
                                                 product_ids, (int)Ni, D, xp, out_item);

    // --- conv up1: users -> items, p1 = lrelu(sage(xu, xp)) ---
    k_zero<<<2048, TB, 0, stream>>>(msg, Ni * HDIM);
    k_zero<<<512,  TB, 0, stream>>>(cnt, Ni);
    k_scatter<<<(unsigned)gScatter, TB, 0, stream>>>(xu, edge_rates, edge_rates + E, E, msg, cnt);
    k_sage<<<(unsigned)gI, TB, 0, stream>>>(msg, cnt, xp, Wl_up1, bl_up1, Wr_up1,
                                            (int)Ni, p1, out_item + 32);

    // --- conv pu1: items -> users, u1 = lrelu(sage(xp, xu)) ---
    k_zero<<<2048, TB, 0, stream>>>(msg, Nu * HDIM);
    k_zero<<<512,  TB, 0, stream>>>(cnt, Nu);
    k_scatter<<<(unsigned)gScatter, TB, 0, stream>>>(xp, edge_rated, edge_rated + E, E, msg, cnt);
    k_sage<<<(unsigned)gU, TB, 0, stream>>>(msg, cnt, xu, Wl_pu1, bl_pu1, Wr_pu1,
                                            (int)Nu, u1, out_user + 32);

    // --- conv up2: u1 -> items, p2 = lrelu(sage(u1, p1)) ---
    k_zero<<<2048, TB, 0, stream>>>(msg, Ni * HDIM);
    k_zero<<<512,  TB, 0, stream>>>(cnt, Ni);
    k_scatter<<<(unsigned)gScatter, TB, 0, stream>>>(u1, edge_rates, edge_rates + E, E, msg, cnt);
    k_sage<<<(unsigned)gI, TB, 0, stream>>>(msg, cnt, p1, Wl_up2, bl_up2, Wr_up2,
                                            (int)Ni, nullptr, out_item + 64);

    // --- conv pu2: p1 -> users, u2 = lrelu(sage(p1, u1)) ---
    k_zero<<<2048, TB, 0, stream>>>(msg, Nu * HDIM);
    k_zero<<<512,  TB, 0, stream>>>(cnt, Nu);
    k_scatter<<<(unsigned)gScatter, TB, 0, stream>>>(p1, edge_rated, edge_rated + E, E, msg, cnt);
    k_sage<<<(unsigned)gU, TB, 0, stream>>>(msg, cnt, u1, Wl_pu2, bl_pu2, Wr_pu2,
                                            (int)Nu, nullptr, out_user + 64);
}